// Point_ContextualEncoder_6047313953084
// MI455X (gfx1250) — compile-verified
//
#include <hip/hip_runtime.h>

// CDNA5 / gfx1250 wave32 WMMA types
typedef float v8f __attribute__((ext_vector_type(8)));
typedef float v2f __attribute__((ext_vector_type(2)));
typedef int   vint4 __attribute__((vector_size(16)));   // builtin's b128 payload type

// gfx1250 async global->LDS copy (ASYNCcnt-tracked), guarded for compile safety.
#if defined(__HIP_DEVICE_COMPILE__) && __has_builtin(__builtin_amdgcn_global_load_async_to_lds_b128)
#define HAVE_ASYNC 1
#else
#define HAVE_ASYNC 0
#endif

__device__ __forceinline__ void wait_async0() {
#if HAVE_ASYNC
#if __has_builtin(__builtin_amdgcn_s_wait_asynccnt)
    __builtin_amdgcn_s_wait_asynccnt(0);
#else
    asm volatile("s_wait_asynccnt 0" ::: "memory");
#endif
#endif
}

// ---------------------------------------------------------------------------
// Weight re-layout (K-permuted for b128 operand loads):
//   Wq[k][c8][half][col(16*NT)][jj(4)] = W[k][8*c8 + 4*half + jj][col]
// (cols >= COUT zero-padded). One float4 per lane = B rows for TWO wmmas:
//   wmma#0 consumes channels {8c+0,1 | 8c+4,5}, wmma#1 {8c+2,3 | 8c+6,7};
// consistent on both A and B sides -> pure K-permutation, exact f32.
// ---------------------------------------------------------------------------
__global__ void reorder_w(const float* __restrict__ W, float* __restrict__ Wp,
                          int L, int CIN, int COUT, int NT16)
{
    const int CIN8 = CIN >> 3;
    const int total = L * CIN8 * 2 * NT16 * 4;
    for (int i = blockIdx.x * blockDim.x + threadIdx.x; i < total;
         i += gridDim.x * blockDim.x) {
        int jj   = i & 3;
        int col  = (i >> 2) % NT16;
        int rest = (i >> 2) / NT16;
        int half = rest & 1;
        int rest2 = rest >> 1;
        int c8   = rest2 % CIN8;
        int k    = rest2 / CIN8;
        float v = 0.f;
        if (col < COUT)
            v = W[((size_t)k * CIN + (8 * c8 + 4 * half + jj)) * COUT + col];
        Wp[i] = v;
    }
}

// elementwise float4 add: o = a + b  (n4 = count of float4)
__global__ void add2_f4(const float4* __restrict__ a, const float4* __restrict__ b,
                        float4* __restrict__ o, size_t n4)
{
    size_t i = blockIdx.x * (size_t)blockDim.x + threadIdx.x;
    if (i < n4) {
        float4 x = a[i], y = b[i];
        x.x += y.x; x.y += y.y; x.z += y.z; x.w += y.w;
        o[i] = x;
    }
}

// ---------------------------------------------------------------------------
// Sparse-conv-as-gather-GEMM using V_WMMA_F32_16X16X4_F32 (exact f32 math).
//   out[m, coloff + c] (stride ldo) = relu( sum_k gather(feat, idx[m,k]) @ W[k] + bias ) (+ resid)
// MT=4 M-tiles per wave; b128 A/B loads each feed 2 wmma; double-buffered LDS;
// async global->LDS staging when available (feat2 forces VGPR path).
// idx == nullptr -> identity gather (1x1 conv). idx value -1 -> zero row.
// ---------------------------------------------------------------------------
template <int CIN, int COUT, int MT>
__global__ void __launch_bounds__(32 * ((COUT + 15) / 16)) spconv_wmma(
    const float* __restrict__ feat, const float* __restrict__ feat2,
    const int* __restrict__ idx, int K, int Nout,
    const float* __restrict__ Wp, const float* __restrict__ bias,
    float* __restrict__ out, int ldo, int coloff,
    const float* __restrict__ resid, int do_relu)
{
    constexpr int NT    = (COUT + 15) / 16;   // N-tiles (waves per block)
    constexpr int NT16  = 16 * NT;
    constexpr int NTH   = 32 * NT;            // threads per block
    constexpr int ROWS  = 16 * MT;            // output points per block
    constexpr int CIN4  = CIN / 4;
    constexpr int CIN8  = CIN / 8;
    constexpr int LDSW  = CIN + 4;            // pad keeps rows 16B-aligned, avoids conflicts
    constexpr bool FULL = (COUT % 16) == 0;
    static_assert(CIN % 8 == 0, "CIN must be a multiple of 8");

    __shared__ float lds[2][ROWS][LDSW];

    const int lane  = threadIdx.x;
    const int ntile = threadIdx.y;
    const int tid   = ntile * 32 + lane;
    const int m0    = blockIdx.x * ROWS;
    const int half  = lane >> 4;
    const int mrow  = lane & 15;
    const int col   = ntile * 16 + (lane & 15);   // 0..NT16-1, always valid in Wq
    const bool colok = FULL || (col < COUT);

    v8f acc[MT];
#pragma unroll
    for (int mt = 0; mt < MT; ++mt) acc[mt] = v8f{};

    // ---- cooperative gather of ROWS rows x CIN floats into LDS buffer `buf` ----
    auto stage = [&](int k, int buf) {
#pragma unroll 1
        for (int e = tid; e < ROWS * CIN4; e += NTH) {
            int m  = e / CIN4;                // constexpr divisor -> shifts
            int q  = e - m * CIN4;
            int gm = m0 + m;
            int r  = -1;
            if (gm < Nout) r = idx ? idx[(size_t)gm * K + k] : gm;
            float* ldst = &lds[buf][m][q * 4];
#if HAVE_ASYNC
            if (!feat2 && r >= 0) {
                // direct global->LDS DMA, no VGPR round-trip (ASYNCcnt)
                __builtin_amdgcn_global_load_async_to_lds_b128(
                    (__attribute__((address_space(1))) vint4*)(feat + (size_t)r * CIN + q * 4),
                    (__attribute__((address_space(3))) vint4*)ldst,
                    0, 0);
                continue;
            }
#endif
            float4 v = make_float4(0.f, 0.f, 0.f, 0.f);
            if (r >= 0) {
                const float4* f4 = reinterpret_cast<const float4*>(feat) + (size_t)r * CIN4 + q;
                v = *f4;
                if (feat2) {                  // fused x_feat + f1_ref on the gather input
                    const float4* g4 = reinterpret_cast<const float4*>(feat2) + (size_t)r * CIN4 + q;
                    float4 w = *g4;
                    v.x += w.x; v.y += w.y; v.z += w.z; v.w += w.w;
                }
            }
            *reinterpret_cast<float4*>(ldst) = v;
        }
    };

    stage(0, 0);
    wait_async0();
    __syncthreads();

    for (int k = 0; k < K; ++k) {
        const int cb = k & 1;
        if (k + 1 < K) stage(k + 1, (k + 1) & 1);   // overlap next gather with this GEMM

        const float* Wk = Wp + (size_t)k * CIN8 * 2 * NT16 * 4;
#pragma unroll
        for (int c8 = 0; c8 < CIN8; ++c8) {
            // one b128 of weights = B fragments for 2 wmmas
            float4 bq = *reinterpret_cast<const float4*>(
                Wk + (((size_t)(c8 * 2 + half) * NT16) + col) * 4);
            v2f b0; b0.x = bq.x; b0.y = bq.y;
            v2f b1; b1.x = bq.z; b1.y = bq.w;
            // one ds b128 per M-tile = A fragments for 2 wmmas
            float4 aq[MT];
#pragma unroll
            for (int mt = 0; mt < MT; ++mt)
                aq[mt] = *reinterpret_cast<const float4*>(
                    &lds[cb][mt * 16 + mrow][c8 * 8 + half * 4]);
#pragma unroll
            for (int mt = 0; mt < MT; ++mt) {   // wmma#0: channels 8c+{0,1|4,5}
                v2f a; a.x = aq[mt].x; a.y = aq[mt].y;
                acc[mt] = __builtin_amdgcn_wmma_f32_16x16x4_f32(
                    false, a, false, b0, (short)0, acc[mt], false, false);
            }
#pragma unroll
            for (int mt = 0; mt < MT; ++mt) {   // wmma#1: channels 8c+{2,3|6,7}
                v2f a; a.x = aq[mt].z; a.y = aq[mt].w;
                acc[mt] = __builtin_amdgcn_wmma_f32_16x16x4_f32(
                    false, a, false, b1, (short)0, acc[mt], false, false);
            }
        }
        wait_async0();     // next tile's DMA must land before any wave touches it
        __syncthreads();
    }

    // ---- epilogue: bias, relu, optional residual, masked store ----
    const float bv = colok ? bias[col] : 0.f;
#pragma unroll
    for (int mt = 0; mt < MT; ++mt) {
#pragma unroll
        for (int v = 0; v < 8; ++v) {
            int gm = m0 + mt * 16 + v + 8 * half;    // C layout: vgpr v -> M = v + 8*half
            float val = acc[mt][v] + bv;
            if (do_relu) val = val > 0.f ? val : 0.f;
            if (gm < Nout && colok) {
                size_t o = (size_t)gm * ldo + coloff + col;
                out[o] = resid ? (val + resid[o]) : val;
            }
        }
    }
}

template <int CIN, int COUT>
static void spconvT(const float* feat, const float* feat2, const int* idx, int K,
                    int Nout, const float* Wp, const float* b,
                    float* out, int ldo, int coloff, const float* resid, int relu,
                    hipStream_t s)
{
    constexpr int MT = 4;
    constexpr int NT = (COUT + 15) / 16;
    dim3 grid((Nout + 16 * MT - 1) / (16 * MT), 1, 1);
    dim3 block(32, NT, 1);
    spconv_wmma<CIN, COUT, MT><<<grid, block, 0, s>>>(feat, feat2, idx, K, Nout,
                                                      Wp, b, out, ldo, coloff, resid, relu);
}

// per-k reordered weight footprint (floats): CIN * 16 * ceil(COUT/16)
static inline size_t wp_per_k(int CIN, int COUT) {
    return (size_t)CIN * 16 * ((COUT + 15) / 16);
}

static void reorder(const float* W, float* Wp, int L, int CIN, int COUT, hipStream_t s) {
    int NT16 = 16 * ((COUT + 15) / 16);
    int total = L * (CIN / 8) * 2 * NT16 * 4;
    int blocks = (total + 255) / 256;
    reorder_w<<<blocks, 256, 0, s>>>(W, Wp, L, CIN, COUT, NT16);
}

extern "C" void kernel_launch(void* const* d_in, const int* in_sizes, int n_in,
                              void* d_out, int out_size, void* d_ws, size_t ws_size,
                              hipStream_t stream)
{
    (void)n_in; (void)out_size;

    const float* x_feat = (const float*)d_in[0];
    const float* f1_ref = (const float*)d_in[1];
    const float* w1   = (const float*)d_in[2];  const float* b1   = (const float*)d_in[3];
    const float* wd   = (const float*)d_in[4];  const float* bd   = (const float*)d_in[5];
    const float* rw00 = (const float*)d_in[6];  const float* rb00 = (const float*)d_in[7];
    const float* rw01 = (const float*)d_in[8];  const float* rb01 = (const float*)d_in[9];
    const float* rw10 = (const float*)d_in[10]; const float* rb10 = (const float*)d_in[11];
    const float* rw11 = (const float*)d_in[12]; const float* rb11 = (const float*)d_in[13];
    const float* rw12 = (const float*)d_in[14]; const float* rb12 = (const float*)d_in[15];
    const float* w4   = (const float*)d_in[16]; const float* b4   = (const float*)d_in[17];
    const int* idx1 = (const int*)d_in[18];
    const int* idxd = (const int*)d_in[19];
    const int* idx2 = (const int*)d_in[20];

    const int N  = in_sizes[0] / 64;     // input points
    const int N2 = in_sizes[19] / 8;     // downsampled points (idx_down is [N2,8])

    // ---- workspace carving (f32) ----
    float* ws   = (float*)d_ws;
    float* h1   = ws;  ws += (size_t)N  * 64;   // conv1 output
    float* outA = ws;  ws += (size_t)N2 * 32;   // block ping
    float* outB = ws;  ws += (size_t)N2 * 32;   // block pong
    float* t0   = ws;  ws += (size_t)N2 * 8;
    float* t1   = ws;  ws += (size_t)N2 * 8;
    float* t2   = ws;  ws += (size_t)N2 * 8;
    // reordered weights
    float* w1p   = ws; ws += 27 * wp_per_k(64, 64);
    float* wdp   = ws; ws += 8  * wp_per_k(64, 32);
    float* rw00p = ws; ws += (size_t)3 * 27 * wp_per_k(32, 8);
    float* rw01p = ws; ws += (size_t)3 * 27 * wp_per_k(8, 16);
    float* rw10p = ws; ws += (size_t)3 * 1  * wp_per_k(32, 8);
    float* rw11p = ws; ws += (size_t)3 * 27 * wp_per_k(8, 8);
    float* rw12p = ws; ws += (size_t)3 * 1  * wp_per_k(8, 16);
    float* w4p   = ws; ws += 27 * wp_per_k(32, 8);
    // optional x_feat + f1_ref buffer (lets conv1 use the async staging path)
    float* xsum  = ws; ws += (size_t)N * 64;
    const bool use_xsum =
        ws_size >= (size_t)(ws - (float*)d_ws) * sizeof(float);

    // ---- one-shot weight re-layouts (tiny; graph-capture safe) ----
    reorder(w1,   w1p,   27,     64, 64, stream);
    reorder(wd,   wdp,   8,      64, 32, stream);
    reorder(rw00, rw00p, 3 * 27, 32, 8,  stream);
    reorder(rw01, rw01p, 3 * 27, 8,  16, stream);
    reorder(rw10, rw10p, 3 * 1,  32, 8,  stream);
    reorder(rw11, rw11p, 3 * 27, 8,  8,  stream);
    reorder(rw12, rw12p, 3 * 1,  8,  16, stream);
    reorder(w4,   w4p,   27,     32, 8,  stream);

    // ---- pipeline ----
    // conv1 + relu : [N,64] -> [N,64]
    if (use_xsum) {
        size_t n4 = (size_t)N * 16;
        add2_f4<<<(unsigned)((n4 + 255) / 256), 256, 0, stream>>>(
            (const float4*)x_feat, (const float4*)f1_ref, (float4*)xsum, n4);
        spconvT<64, 64>(xsum, nullptr, idx1, 27, N, w1p, b1, h1, 64, 0, nullptr, 1, stream);
    } else {
        spconvT<64, 64>(x_feat, f1_ref, idx1, 27, N, w1p, b1, h1, 64, 0, nullptr, 1, stream);
    }
    // down k2/s2 + relu : [N,64] -> [N2,32]
    spconvT<64, 32>(h1, nullptr, idxd, 8, N2, wdp, bd, outA, 32, 0, nullptr, 1, stream);

    // 3 InceptionResNet blocks on [N2,32]
    for (int i = 0; i < 3; ++i) {
        float* cur = (i & 1) ? outB : outA;
        float* nxt = (i & 1) ? outA : outB;
        const float* W00 = rw00p + (size_t)i * 27 * wp_per_k(32, 8);
        const float* W01 = rw01p + (size_t)i * 27 * wp_per_k(8, 16);
        const float* W10 = rw10p + (size_t)i * 1  * wp_per_k(32, 8);
        const float* W11 = rw11p + (size_t)i * 27 * wp_per_k(8, 8);
        const float* W12 = rw12p + (size_t)i * 1  * wp_per_k(8, 16);
        const float* B00 = rb00 + (size_t)i * 8;
        const float* B01 = rb01 + (size_t)i * 16;
        const float* B10 = rb10 + (size_t)i * 8;
        const float* B11 = rb11 + (size_t)i * 8;
        const float* B12 = rb12 + (size_t)i * 16;

        // branch 0: relu(conv 32->8), then conv 8->16 into cols [0:16) with residual += cur
        spconvT<32, 8 >(cur, nullptr, idx2, 27, N2, W00, B00, t0,  8,  0,  nullptr, 1, stream);
        spconvT<8,  16>(t0,  nullptr, idx2, 27, N2, W01, B01, nxt, 32, 0,  cur,     0, stream);
        // branch 1: relu(1x1 32->8), relu(conv 8->8), 1x1 8->16 into cols [16:32) with residual += cur
        spconvT<32, 8 >(cur, nullptr, nullptr, 1, N2, W10, B10, t1,  8,  0,  nullptr, 1, stream);
        spconvT<8,  8 >(t1,  nullptr, idx2,  27, N2, W11, B11, t2,  8,  0,  nullptr, 1, stream);
        spconvT<8,  16>(t2,  nullptr, nullptr, 1, N2, W12, B12, nxt, 32, 16, cur,     0, stream);
    }

    // after i=0: A->B, i=1: B->A, i=2: A->B  => final features live in outB
    spconvT<32, 8>(outB, nullptr, idx2, 27, N2, w4p, b4, (float*)d_out, 8, 0, nullptr, 0, stream);
}